// TemporalDFFN_76776835383440
// MI455X (gfx1250) — compile-verified
//
#include <hip/hip_runtime.h>

// Spectral filter y = irfft(rfft(x) * exp(clip(W,-2,2))) with T=32.
// The op is a per-channel CIRCULANT: M_c[t][t'] = h_c[(t-t') mod 32],
// h_c[d] = (1/32)(s0 + (-1)^d s16 + 2*sum_{k=1..15} s_k cos(2pi k d/32)).
// Batch-major tiles at fixed channel -> one fp32 WMMA GEMM per 16x32 tile.
// All trig comes from a compile-time cos(2pi n/32) table (no libm), and the
// per-lane B-fragments live in an LDS table read with immediate-offset
// loads once, into registers, before the main loop.

typedef float v2f __attribute__((ext_vector_type(2)));
typedef float v4f __attribute__((ext_vector_type(4)));
typedef float v8f __attribute__((ext_vector_type(8)));

#define TLEN 32
#define NBINS 17
#define NCH 512
#define NBATCH 4096
#define NWAVES 8
#define BATCHES_PER_BLOCK 256
#define ITERS (BATCHES_PER_BLOCK / (NWAVES * 16))   // 2
#define BT_ST 36   // Btab per-lane stride (floats): 16B-aligned, conflict-free
#define ST 36      // output staging row stride (floats)

// cos(2*pi*n/32), n = 0..31 (exact fp32 constants)
__constant__ float kCosT[32] = {
     1.0f,          0.98078528f,  0.92387953f,  0.83146961f,
     0.70710678f,   0.55557023f,  0.38268343f,  0.19509032f,
     0.0f,         -0.19509032f, -0.38268343f, -0.55557023f,
    -0.70710678f,  -0.83146961f, -0.92387953f, -0.98078528f,
    -1.0f,         -0.98078528f, -0.92387953f, -0.83146961f,
    -0.70710678f,  -0.55557023f, -0.38268343f, -0.19509032f,
     0.0f,          0.19509032f,  0.38268343f,  0.55557023f,
     0.70710678f,   0.83146961f,  0.92387953f,  0.98078528f
};

__global__ __launch_bounds__(256) void spectral_circulant_kernel(
    const float* __restrict__ x,       // (B, C, 32)
    const float* __restrict__ W,       // (C, 17)
    float* __restrict__ y)             // (B, C, 32)
{
    __shared__ float Sc[NBINS];               // exp(clip(W[c,:]))
    __shared__ float Hc[TLEN];                // circulant taps h[d]
    __shared__ float Btab[32 * BT_ST];        // per-lane B-fragments (32 floats/lane)
    __shared__ float Stg[NWAVES * 16 * ST];   // per-wave output staging

    const int tid = threadIdx.x;
    const int c   = blockIdx.x;                        // channel
    const int b0  = blockIdx.y * BATCHES_PER_BLOCK;    // batch base

    // ---- Phase 0a: per-bin scales (fast hw exp; ~1e-7 rel err) ----
    if (tid < NBINS) {
        float wv = W[c * NBINS + tid];
        wv = fminf(fmaxf(wv, -2.0f), 2.0f);
        Sc[tid] = __expf(wv);
    }
    __syncthreads();

    // ---- Phase 0b: circulant taps via compile-time cosine table ----
    if (tid < TLEN) {
        float acc = Sc[0] + ((tid & 1) ? -Sc[16] : Sc[16]);
        #pragma unroll
        for (int k = 1; k < 16; ++k)
            acc += 2.0f * Sc[k] * kCosT[(k * tid) & 31];
        Hc[tid] = acc * (1.0f / 32.0f);
    }
    __syncthreads();

    // ---- Phase 0c: per-lane B-fragment table (identical for all 8 waves) ----
    // Fragment element (lane, nt, kk, d) <-> M[K = 4kk+2hi+d][N = nt*16+m]
    //   = h[(K - N) & 31];  stored at Btab[lane*BT_ST + nt*16 + 2*kk + d].
    for (int e = tid; e < 1024; e += 256) {
        const int ln = e >> 5;                 // lane
        const int r  = e & 31;                 // nt*16 + 2*kk + d
        const int lm = ln & 15, lhi = ln >> 4;
        const int nt = r >> 4;
        const int kk = (r & 15) >> 1;
        const int d  = r & 1;
        const int n  = nt * 16 + lm;
        const int K  = 4 * kk + 2 * lhi + d;
        Btab[ln * BT_ST + r] = Hc[(K - n) & 31];
    }
    __syncthreads();

    const int wave = tid >> 5;
    const int lane = tid & 31;
    const int m    = lane & 15;
    const int hi   = lane >> 4;

    // ---- Loop-invariant B-fragments: contiguous, immediate offsets ----
    const float* btl = &Btab[lane * BT_ST];
    v2f bfr[2][8];
    #pragma unroll
    for (int nt = 0; nt < 2; ++nt)
        #pragma unroll
        for (int kk = 0; kk < 8; ++kk)
            bfr[nt][kk] = *(const v2f*)(btl + nt * 16 + 2 * kk);

    float* stg = &Stg[wave * (16 * ST)];
    const size_t rowstride = (size_t)NCH * TLEN;   // 16384 floats between batches

    for (int it = 0; it < ITERS; ++it) {
        const int bt0 = b0 + it * (NWAVES * 16) + wave * 16;

        // A-fragments of x: component d <-> K = 4kk + 2hi + d, row M = m
        const float* xrow = x + (size_t)(bt0 + m) * rowstride + (size_t)c * TLEN;
        v2f a[8];
        #pragma unroll
        for (int kk = 0; kk < 8; ++kk)
            a[kk] = *(const v2f*)(xrow + 4 * kk + 2 * hi);

        // ---- single GEMM: Y_tile(16x32) = X_tile(16x32) * M(32x32) ----
        v8f u[2];
        #pragma unroll
        for (int nt = 0; nt < 2; ++nt) {
            v8f acc = {};
            #pragma unroll
            for (int kk = 0; kk < 8; ++kk)
                acc = __builtin_amdgcn_wmma_f32_16x16x4_f32(
                          false, a[kk], false, bfr[nt][kk], (short)0, acc, false, false);
            u[nt] = acc;
        }

        // ---- stage C/D layout -> row-major LDS (same-wave DS, in-order) ----
        #pragma unroll
        for (int nt = 0; nt < 2; ++nt) {
            const int n = nt * 16 + m;
            #pragma unroll
            for (int v = 0; v < 8; ++v)
                stg[(v + 8 * hi) * ST + n] = u[nt][v];
        }
        // ---- b128 stores: lane pair (l, l+16) covers one 128B batch row ----
        float* yrow = y + (size_t)(bt0 + m) * rowstride + (size_t)c * TLEN + hi * 16;
        #pragma unroll
        for (int q = 0; q < 4; ++q) {
            v4f val = *(const v4f*)(&stg[m * ST + hi * 16 + q * 4]);
            *(v4f*)(yrow + q * 4) = val;
        }
    }
}

extern "C" void kernel_launch(void* const* d_in, const int* in_sizes, int n_in,
                              void* d_out, int out_size, void* d_ws, size_t ws_size,
                              hipStream_t stream) {
    (void)in_sizes; (void)n_in; (void)d_ws; (void)ws_size; (void)out_size;
    const float* x = (const float*)d_in[0];   // (4096, 512, 32) fp32
    const float* W = (const float*)d_in[1];   // (1, 512, 17)  fp32
    float* y = (float*)d_out;                 // (4096, 512, 32) fp32

    dim3 grid(NCH, NBATCH / BATCHES_PER_BLOCK);   // (512, 16)
    dim3 block(256);                              // 8 waves
    hipLaunchKernelGGL(spectral_circulant_kernel, grid, block, 0, stream, x, W, y);
}